// NGCF_24816321037031
// MI455X (gfx1250) — compile-verified
//
#include <hip/hip_runtime.h>
#include <hip/hip_bf16.h>
#include <math.h>

#define N_USER  100000
#define N_ITEM  50000
#define N_NODES 150000
#define D       64
#define NNZ     2400000
#define BATCH   4096
#define NEG_SLOPE 0.2f
#define EPS     1e-12f

typedef __attribute__((ext_vector_type(2))) float v2f;
typedef __attribute__((ext_vector_type(8))) float v8f;

// ---------------------------------------------------------------------------
// ego0 = concat(user_emb, item_emb)
// ---------------------------------------------------------------------------
__global__ void ngcf_concat(const float* __restrict__ ue,
                            const float* __restrict__ ie,
                            float* __restrict__ ego) {
    int i = blockIdx.x * blockDim.x + threadIdx.x;
    const int total = N_NODES * D;
    if (i < total) {
        const int split = N_USER * D;
        ego[i] = (i < split) ? ue[i] : ie[i - split];
    }
}

__global__ void ngcf_zero(float* __restrict__ p, int n) {
    int i = blockIdx.x * blockDim.x + threadIdx.x;
    if (i < n) p[i] = 0.0f;
}

// ---------------------------------------------------------------------------
// SpMM: side[row] += val * x[col]   (COO scatter, one wave per edge,
// each lane handles 2 of the 64 features; fire-and-forget f32 atomics)
// ---------------------------------------------------------------------------
__global__ void ngcf_spmm(const int* __restrict__ rows,
                          const int* __restrict__ cols,
                          const float* __restrict__ vals,
                          const float* __restrict__ x,
                          float* __restrict__ y) {
    int t = blockIdx.x * blockDim.x + threadIdx.x;
    int e = t >> 5;
    int lane = t & 31;
    if (e >= NNZ) return;
    int r = rows[e];
    int c = cols[e];
    float v = vals[e];
    float2 m = *(const float2*)(x + (size_t)c * D + lane * 2);
    float* yp = y + (size_t)r * D + lane * 2;
    atomicAdd(yp + 0, v * m.x);
    atomicAdd(yp + 1, v * m.y);
}

// ---------------------------------------------------------------------------
// Fused layer:
//   h        = leaky_relu( side@Wgc + bgc + (ego*side)@Wbi + bbi )
//   ego_next = h            (propagated unnormalized, per reference)
//   norm_out = h / max(||h||_2, eps)
// One wave computes a 16-row tile via v_wmma_f32_16x16x4_f32.
//
// Weights are staged in LDS pre-swizzled to B-fragment layout:
//   sF[m][ks][t][lane] = { W[ks*4 + (lane>>4)*2    ][t*16 + (lane&15)],
//                          W[ks*4 + (lane>>4)*2 + 1][t*16 + (lane&15)] }
// so each WMMA B operand is ONE aligned ds_load_b64 (lane L -> dwords
// 2L,2L+1: all 64 banks, conflict-free, no register-pair fix-up moves).
// ---------------------------------------------------------------------------
__global__ void __launch_bounds__(256)
ngcf_layer(const float* __restrict__ side,
           const float* __restrict__ ego,
           const float* __restrict__ Wgc,
           const float* __restrict__ bgc,
           const float* __restrict__ Wbi,
           const float* __restrict__ bbi,
           float* __restrict__ ego_next,
           float* __restrict__ norm_out) {
    __shared__ v2f   sF[2][16][4][32];   // 32 KB
    __shared__ float sB[D];

    const int tid = threadIdx.x;         // 256 threads = 8 waves
    // stage both 64x64 weight matrices (swizzled) + combined bias into LDS
    for (int i = tid; i < D * D; i += 256) {
        const int r = i >> 6, c = i & 63;
        const int ks   = r >> 2;
        const int koff = r & 3;
        const int hf   = koff >> 1;      // which 16-lane half
        const int j    = koff & 1;       // element within the float2
        const int t    = c >> 4;
        const int ln   = hf * 16 + (c & 15);
        ((float*)&sF[0][ks][t][ln])[j] = Wgc[i];
        ((float*)&sF[1][ks][t][ln])[j] = Wbi[i];
    }
    if (tid < D) sB[tid] = bgc[tid] + bbi[tid];
    __syncthreads();

    const int wave  = tid >> 5;
    const int lane  = tid & 31;
    const int blk16 = blockIdx.x * 8 + wave;          // 16-row tile index
    if (blk16 >= N_NODES / 16) return;                // wave-uniform guard

    const int half = lane >> 4;                       // 0 | 1
    const int l16  = lane & 15;
    const size_t row = (size_t)blk16 * 16 + l16;      // A rows (M = lane&15)
    const float* sp = side + row * D;
    const float* ep = ego  + row * D;

    // 4 N-tiles of 16 cols; accumulators seeded with combined bias
    v8f acc[4];
    for (int t = 0; t < 4; ++t) {
        float b = sB[t * 16 + l16];
        for (int j = 0; j < 8; ++j) acc[t][j] = b;
    }

    for (int ks = 0; ks < 16; ++ks) {                 // K = 64 in steps of 4
        const int k0 = ks * 4 + half * 2;
        v2f a_side = *(const v2f*)(sp + k0);
        v2f fe     = *(const v2f*)(ep + k0);
        v2f a_prod = a_side * fe;
        for (int t = 0; t < 4; ++t) {
            v2f b_gc = sF[0][ks][t][lane];
            acc[t] = __builtin_amdgcn_wmma_f32_16x16x4_f32(
                false, a_side, false, b_gc, (short)0, acc[t], false, false);
        }
        for (int t = 0; t < 4; ++t) {
            v2f b_bi = sF[1][ks][t][lane];
            acc[t] = __builtin_amdgcn_wmma_f32_16x16x4_f32(
                false, a_prod, false, b_bi, (short)0, acc[t], false, false);
        }
    }

    // leaky-relu + per-row sum of squares (row m = j + 8*half lives in one
    // 16-lane half, spread across lanes 0..15 of that half over 4 N-tiles)
    float ss[8];
    for (int j = 0; j < 8; ++j) ss[j] = 0.0f;
    for (int t = 0; t < 4; ++t)
        for (int j = 0; j < 8; ++j) {
            float x = acc[t][j];
            x = (x > 0.0f) ? x : NEG_SLOPE * x;
            acc[t][j] = x;
            ss[j] += x * x;
        }
    for (int m = 1; m < 16; m <<= 1)
        for (int j = 0; j < 8; ++j)
            ss[j] += __shfl_xor(ss[j], m, 32);
    float scale[8];
    for (int j = 0; j < 8; ++j)
        scale[j] = 1.0f / fmaxf(sqrtf(ss[j]), EPS);

    for (int t = 0; t < 4; ++t)
        for (int j = 0; j < 8; ++j) {
            size_t r = (size_t)blk16 * 16 + j + 8 * half;
            int    c = t * 16 + l16;
            ego_next[r * D + c] = acc[t][j];
            norm_out[r * D + c] = acc[t][j] * scale[j];
        }
}

// ---------------------------------------------------------------------------
// Final gather: out[g][b][0:64]=ego0, [64:128]=n1, [128:192]=n2, [192:256]=n3
// ---------------------------------------------------------------------------
__global__ void ngcf_gather(const int* __restrict__ user,
                            const int* __restrict__ item_i,
                            const int* __restrict__ item_j,
                            const float* __restrict__ ue,
                            const float* __restrict__ ie,
                            const float* __restrict__ n1,
                            const float* __restrict__ n2,
                            const float* __restrict__ n3,
                            float* __restrict__ out) {
    int t = blockIdx.x * blockDim.x + threadIdx.x;   // over 3*BATCH*D
    if (t >= 3 * BATCH * D) return;
    int d = t & 63;
    int b = (t >> 6) & (BATCH - 1);
    int g = t >> 18;                                  // BATCH*D = 2^18
    int node = (g == 0) ? user[b]
             : (g == 1) ? N_USER + item_i[b]
                        : N_USER + item_j[b];
    size_t nd = (size_t)node * D + d;
    float e0 = (node < N_USER) ? ue[nd] : ie[(size_t)(node - N_USER) * D + d];
    size_t ob = ((size_t)g * BATCH + b) * (4 * D);
    out[ob +       d] = e0;
    out[ob +  64 + d] = n1[nd];
    out[ob + 128 + d] = n2[nd];
    out[ob + 192 + d] = n3[nd];
}

// ---------------------------------------------------------------------------
extern "C" void kernel_launch(void* const* d_in, const int* in_sizes, int n_in,
                              void* d_out, int out_size, void* d_ws, size_t ws_size,
                              hipStream_t stream) {
    const int*   user    = (const int*)d_in[0];
    const int*   item_i  = (const int*)d_in[1];
    const int*   item_j  = (const int*)d_in[2];
    const int*   adj_r   = (const int*)d_in[3];
    const int*   adj_c   = (const int*)d_in[4];
    const float* adj_v   = (const float*)d_in[5];
    const float* user_e  = (const float*)d_in[6];
    const float* item_e  = (const float*)d_in[7];
    const float* W_gc    = (const float*)d_in[8];
    const float* b_gc    = (const float*)d_in[9];
    const float* W_bi    = (const float*)d_in[10];
    const float* b_bi    = (const float*)d_in[11];
    float* out = (float*)d_out;

    const size_t ND = (size_t)N_NODES * D;
    float* ws    = (float*)d_ws;
    float* ego_a = ws;
    float* ego_b = ws + ND;
    float* side  = ws + 2 * ND;
    float* norm0 = ws + 3 * ND;
    float* norm1 = ws + 4 * ND;
    float* norm2 = ws + 5 * ND;
    float* norms[3] = {norm0, norm1, norm2};

    const int totalND = N_NODES * D;
    ngcf_concat<<<(totalND + 255) / 256, 256, 0, stream>>>(user_e, item_e, ego_a);

    float* cur = ego_a;
    float* nxt = ego_b;
    for (int k = 0; k < 3; ++k) {
        ngcf_zero<<<(totalND + 255) / 256, 256, 0, stream>>>(side, totalND);
        ngcf_spmm<<<(NNZ * 32) / 256, 256, 0, stream>>>(adj_r, adj_c, adj_v, cur, side);
        const int tiles = N_NODES / 16;            // 9375
        ngcf_layer<<<(tiles + 7) / 8, 256, 0, stream>>>(
            side, cur,
            W_gc + (size_t)k * D * D, b_gc + (size_t)k * D,
            W_bi + (size_t)k * D * D, b_bi + (size_t)k * D,
            nxt, norms[k]);
        float* tmp = cur; cur = nxt; nxt = tmp;
    }

    ngcf_gather<<<(3 * BATCH * D + 255) / 256, 256, 0, stream>>>(
        user, item_i, item_j, user_e, item_e, norm0, norm1, norm2, out);
}